// ChamferDistanceCriterion_29781303231230
// MI455X (gfx1250) — compile-verified
//
#include <hip/hip_runtime.h>
#include <hip/hip_bf16.h>
#include <math.h>

typedef __attribute__((ext_vector_type(2))) float v2f;
typedef __attribute__((ext_vector_type(8))) float v8f;

#define BB 64
#define TT 64
#define VV 8192

__global__ void zero_out_kernel(float* __restrict__ out, int n) {
    int i = blockIdx.x * blockDim.x + threadIdx.x;
    if (i < n) out[i] = 0.0f;
}

// ---------------------------------------------------------------------------
// Kernel A: per-(b,t) row softmax statistics (rowmax, sum of exp).
// One workgroup (256 threads = 8 wave32) per row of 8192 f32 logits.
// Memory-bound: single coalesced pass (b128 loads). The sum-of-exp reduction
// runs on the matrix unit via V_WMMA_F32_16X16X4_F32 with an all-ones B
// matrix (D[m][n] = sum_k A[m][k]), co-executing with the v_exp_f32 stream.
// ---------------------------------------------------------------------------
__global__ __launch_bounds__(256) void softmax_stats_kernel(
    const float* __restrict__ logits,
    float* __restrict__ rowmax,
    float* __restrict__ rowsum) {
    const int row = blockIdx.x;          // 0..B*T-1
    const int tid = threadIdx.x;         // 0..255
    const float* base = logits + (size_t)row * VV;

    // Load 32 elements per thread: 8 chunks of contiguous float4 (coalesced).
    float4 v[8];
#pragma unroll
    for (int c = 0; c < 8; ++c) {
        v[c] = ((const float4*)(base + c * 1024))[tid];
    }

    // Row max: per-thread max then LDS tree.
    float mx = -INFINITY;
#pragma unroll
    for (int c = 0; c < 8; ++c) {
        mx = fmaxf(mx, fmaxf(fmaxf(v[c].x, v[c].y), fmaxf(v[c].z, v[c].w)));
    }
    __shared__ float red[256];
    __shared__ float wred[8];
    red[tid] = mx;
    __syncthreads();
    for (int s = 128; s > 0; s >>= 1) {
        if (tid < s) red[tid] = fmaxf(red[tid], red[tid + s]);
        __syncthreads();
    }
    const float m = red[0];
    __syncthreads();

    // exp partials: one partial per float4 -> 8 f32 per lane.
    float p[8];
#pragma unroll
    for (int c = 0; c < 8; ++c) {
        p[c] = __expf(v[c].x - m) + __expf(v[c].y - m) +
               __expf(v[c].z - m) + __expf(v[c].w - m);
    }

    // Wave-wide sum on the matrix unit: 4 accumulating f32 WMMAs consume all
    // 8 partials x 32 lanes. With B == ones, D[m][n] = sum_k A[m][k]; summing
    // the 8 D VGPRs per lane leaves lane0 = sum over M=0..7, lane16 = M=8..15.
    v2f ones; ones[0] = 1.0f; ones[1] = 1.0f;
    v8f acc = {};
    v2f a;
    a[0] = p[0]; a[1] = p[1];
    acc = __builtin_amdgcn_wmma_f32_16x16x4_f32(false, a, false, ones, (short)0, acc, false, false);
    a[0] = p[2]; a[1] = p[3];
    acc = __builtin_amdgcn_wmma_f32_16x16x4_f32(false, a, false, ones, (short)0, acc, false, false);
    a[0] = p[4]; a[1] = p[5];
    acc = __builtin_amdgcn_wmma_f32_16x16x4_f32(false, a, false, ones, (short)0, acc, false, false);
    a[0] = p[6]; a[1] = p[7];
    acc = __builtin_amdgcn_wmma_f32_16x16x4_f32(false, a, false, ones, (short)0, acc, false, false);

    float vsum = acc[0] + acc[1] + acc[2] + acc[3] + acc[4] + acc[5] + acc[6] + acc[7];
    float wtot = __shfl(vsum, 0, 32) + __shfl(vsum, 16, 32);

    const int lane = tid & 31, wid = tid >> 5;
    if (lane == 0) wred[wid] = wtot;
    __syncthreads();
    if (tid == 0) {
        float s = 0.0f;
#pragma unroll
        for (int w = 0; w < 8; ++w) s += wred[w];
        rowmax[row] = m;
        rowsum[row] = s;
    }
}

// ---------------------------------------------------------------------------
// Kernel B: per-batch chamfer + eos loss using the analytic collapse of the
// one-hot GEMMs. One workgroup (64 threads) per batch:
//   G[i][j] = probs[b, i, targets[b,j]]   (64x64 gather, LDS-staged)
//   dir1_n  = -log(clip(max_i mask_i * G[i][n], eps, hi))            (mask_n)
//   dir2_n  = -log(eps)*(1 - p0_n) + (log(eps)-log(hi)) * max_{mask_j} G[n][j]
//   eos     = masked BCE on probs[:,:,0]
// ---------------------------------------------------------------------------
__global__ __launch_bounds__(64) void chamfer_batch_kernel(
    const float* __restrict__ logits,
    const long long* __restrict__ targets,
    const float* __restrict__ rowmax,
    const float* __restrict__ rowsum,
    float* __restrict__ out) {
    const int b = blockIdx.x;
    const int j = threadIdx.x;           // 0..63

    __shared__ float mxs[TT], isms[TT], p0s[TT];
    __shared__ int   tgt[TT], msk[TT];
    __shared__ float G[TT][TT + 1];      // stride 65: conflict-free row reads
    __shared__ float r0[TT], r1[TT], r2[TT], r3[TT], r4[TT];

    const int r = b * TT + j;
    const long long t64 = targets[r];
    const int tj = (int)t64;
    const int mask = (tj != 0 && tj != VV) ? 1 : 0;   // eos_head

    const float mx  = rowmax[r];
    const float ism = 1.0f / rowsum[r];
    const float p0  = __expf(logits[(size_t)r * VV] - mx) * ism;  // eos prob

    mxs[j] = mx; isms[j] = ism; p0s[j] = p0; tgt[j] = tj; msk[j] = mask;
    __syncthreads();

    // Column j of G (only needed for masked j; t_j in [1,8191] then, so the
    // gather index is always in range).
    float colmax = 0.0f;
    if (mask) {
        const float* lb = logits + (size_t)b * TT * VV + tj;
#pragma unroll 4
        for (int i = 0; i < TT; ++i) {
            const float g = __expf(lb[(size_t)i * VV] - mxs[i]) * isms[i];
            G[i][j] = g;
            if (msk[i]) colmax = fmaxf(colmax, g);
        }
    }
    __syncthreads();

    const float EPSf    = 1e-8f;
    const float HI      = 0.99991810f;       // 1 - (8191-1)*1e-8
    const float LOG_EPS = -18.420680744f;    // log(1e-8)
    const float LOG_HI  = -8.1903356e-5f;    // log(HI)

    float loss = 0.0f;
    if (mask) {
        // direction 1: knn_ce(one_hot, xs)
        const float c = fminf(fmaxf(colmax, EPSf), HI);
        loss += -__logf(c);
        // direction 2: knn_ce(xs, one_hot)
        float rowm = 0.0f;
        for (int mm = 0; mm < TT; ++mm)
            if (msk[mm]) rowm = fmaxf(rowm, G[j][mm]);
        const float S = 1.0f - p0;           // sum of xs row (mask_n == 1)
        loss += -LOG_EPS * S + (LOG_EPS - LOG_HI) * rowm;
    }

    // eos BCE: et = !eos_head
    const float bce = mask ? -__logf(1.0f - p0) : -__logf(p0);
    const float ep  = (tj == 0) ? 1.0f : 0.0f;
    const float eh  = mask ? 1.0f : 0.0f;

    r0[j] = loss;
    r1[j] = bce * ep;  r2[j] = ep;
    r3[j] = bce * eh;  r4[j] = eh;
    __syncthreads();
    for (int s = 32; s > 0; s >>= 1) {
        if (j < s) {
            r0[j] += r0[j + s];
            r1[j] += r1[j + s];
            r2[j] += r2[j + s];
            r3[j] += r3[j + s];
            r4[j] += r4[j + s];
        }
        __syncthreads();
    }
    if (j == 0) {
        atomicAdd(out + 0, r0[0]);
        const float be = 0.5f * r1[0] / (r2[0] + EPSf) +
                         0.5f * r3[0] / (r4[0] + EPSf);
        atomicAdd(out + 1, be * (1.0f / (float)BB));
    }
}

extern "C" void kernel_launch(void* const* d_in, const int* in_sizes, int n_in,
                              void* d_out, int out_size, void* d_ws, size_t ws_size,
                              hipStream_t stream) {
    (void)in_sizes; (void)n_in; (void)ws_size;
    const float*     logits  = (const float*)d_in[0];
    const long long* targets = (const long long*)d_in[1];  // jnp.int64
    float* out = (float*)d_out;

    float* rowmax = (float*)d_ws;           // B*T f32
    float* rowsum = rowmax + BB * TT;       // B*T f32

    zero_out_kernel<<<1, 64, 0, stream>>>(out, out_size);
    softmax_stats_kernel<<<BB * TT, 256, 0, stream>>>(logits, rowmax, rowsum);
    chamfer_batch_kernel<<<BB, 64, 0, stream>>>(logits, targets, rowmax, rowsum, out);
}